// SparseActivationEnforcer_18700287607417
// MI455X (gfx1250) — compile-verified
//
#include <hip/hip_runtime.h>

// ---------------------------------------------------------------------------
// SparseActivationEnforcer for MI455X (gfx1250).
// Streaming 2:4-sparsity mask in fp16-abs domain; row 0 pass-through.
// Memory-bound: 268 MB traffic -> ~11.5 us floor at 23.3 TB/s.
// Data path: async global->LDS b128 staging (ASYNCcnt, double-buffered,
//            NT load hint) + non-temporal b128 stores.
// ---------------------------------------------------------------------------

typedef float v4f __attribute__((ext_vector_type(4)));
typedef int   v4i __attribute__((ext_vector_type(4)));

// Pointer flavors for the async-copy builtin (param types per hipcc diag:
// pointee is int4; global side is AS(1) ("__device__"), LDS side AS(3)).
typedef __attribute__((address_space(1))) v4i* as1_v4i_p;
typedef __attribute__((address_space(3))) v4i* as3_v4i_p;

#define BLOCK_THREADS 256
#define GROUPS_PER_TILE 256   // one 16B group per thread per tile

// CPol: TH field in bits [2:0]; 1 = NT (stream, don't retain in caches).
#define CPOL_TH_NT 1

// ---- fp16 |value| as integer bits (monotonic ordering for non-NaN) --------
__device__ __forceinline__ unsigned habs16(float f) {
  _Float16 h = (_Float16)f;            // v_cvt_f16_f32, RNE (matches jnp astype)
  unsigned short b;
  __builtin_memcpy(&b, &h, 2);
  return (unsigned)(b & 0x7FFFu);
}

// ---- 2:4 mask of one group; 'passthrough' for the head row ----------------
__device__ __forceinline__ v4f mask24(v4f v, bool passthrough) {
  const unsigned a0 = habs16(v.x), a1 = habs16(v.y);
  const unsigned a2 = habs16(v.z), a3 = habs16(v.w);
  // rank_i = #{j<i : a_j >= a_i} + #{j>i : a_j > a_i}  (stable-sort tiebreak)
  const int r0 = (a1 >  a0) + (a2 >  a0) + (a3 >  a0);
  const int r1 = (a0 >= a1) + (a2 >  a1) + (a3 >  a1);
  const int r2 = (a0 >= a2) + (a1 >= a2) + (a3 >  a2);
  const int r3 = (a0 >= a3) + (a1 >= a3) + (a2 >= a3);
  v4f r;
  r.x = (passthrough || ((r0 < 2) && (a0 != 0u))) ? v.x : 0.0f;
  r.y = (passthrough || ((r1 < 2) && (a1 != 0u))) ? v.y : 0.0f;
  r.z = (passthrough || ((r2 < 2) && (a2 != 0u))) ? v.z : 0.0f;
  r.w = (passthrough || ((r3 < 2) && (a3 != 0u))) ? v.w : 0.0f;
  return r;
}

// ---- CDNA5 async global->LDS 16B copy (per-lane VGPR addresses) -----------
__device__ __forceinline__ void async_load16(const float* gp, v4f* lp) {
#if defined(__HIP_DEVICE_COMPILE__)
#if __has_builtin(__builtin_amdgcn_global_load_async_to_lds_b128)
  __builtin_amdgcn_global_load_async_to_lds_b128(
      (as1_v4i_p)(void*)const_cast<float*>(gp),
      (as3_v4i_p)(void*)lp,
      /*offset=*/0, /*cpol=*/CPOL_TH_NT);
#else
  // Fallback: raw mnemonic per CDNA5 ISA.
  // VDST = 32-bit LDS byte offset, VADDR = 64-bit global address (GV mode).
  unsigned lds_off = (unsigned)(unsigned long long)
      ((__attribute__((address_space(3))) void*)((void*)lp));
  asm volatile("global_load_async_to_lds_b128 %0, %1, off th:TH_LOAD_NT"
               :: "v"(lds_off), "v"(gp) : "memory");
#endif
#endif
}

// Wave-level wait on outstanding async copies. asm + memory clobber so the
// subsequent LDS read cannot be hoisted above the wait.
#define WAIT_ASYNC(n) asm volatile("s_wait_asynccnt %0" :: "n"(n) : "memory")

// ---------------------------------------------------------------------------
// Main pipelined kernel: double-buffered async staging through LDS.
// Each thread owns one private 16B LDS slot per buffer -> no barriers needed,
// only the per-wave ASYNCcnt wait.
// ---------------------------------------------------------------------------
__global__ __launch_bounds__(BLOCK_THREADS)
void sae_main(const float* __restrict__ x, float* __restrict__ out,
              long long ntiles, long long headGroups) {
  __shared__ v4f tile[2][BLOCK_THREADS];   // 8 KB of the 320 KB WGP LDS
  const int tid = threadIdx.x;
  const long long stride = gridDim.x;

  long long t = blockIdx.x;
  if (t < ntiles) {
    async_load16(x + ((t * GROUPS_PER_TILE + tid) << 2), &tile[0][tid]);
  }

  int buf = 0;
  for (; t < ntiles; t += stride) {
    const long long tn = t + stride;
    if (tn < ntiles) {
      // Prefetch next tile into the other buffer, then wait for current tile.
      async_load16(x + ((tn * GROUPS_PER_TILE + tid) << 2), &tile[buf ^ 1][tid]);
      WAIT_ASYNC(1);
    } else {
      WAIT_ASYNC(0);
    }

    const long long g = t * GROUPS_PER_TILE + tid;   // group index
    v4f v = tile[buf][tid];                          // ds_load_b128
    v4f r = mask24(v, g < headGroups);
    // Output never re-read; 256 MB stream > 192 MB L2 -> non-temporal store.
    __builtin_nontemporal_store(r, (v4f*)(out + (g << 2)));
    buf ^= 1;
  }
}

// ---------------------------------------------------------------------------
// Tail kernel (generic guard; unused for the 8192x4096 shape, which divides
// into 32768 exact tiles).
// ---------------------------------------------------------------------------
__global__ __launch_bounds__(BLOCK_THREADS)
void sae_tail(const float* __restrict__ x, float* __restrict__ out,
              long long startGroup, long long totalGroups, long long headGroups) {
  const long long g = startGroup +
      (long long)blockIdx.x * blockDim.x + threadIdx.x;
  if (g >= totalGroups) return;
  v4f v = *(const v4f*)(x + (g << 2));
  v4f r = mask24(v, g < headGroups);
  __builtin_nontemporal_store(r, (v4f*)(out + (g << 2)));
}

// ---------------------------------------------------------------------------
extern "C" void kernel_launch(void* const* d_in, const int* in_sizes, int n_in,
                              void* d_out, int out_size, void* d_ws, size_t ws_size,
                              hipStream_t stream) {
  const float* x = (const float*)d_in[0];
  float* out = (float*)d_out;

  const long long n = (long long)in_sizes[0];       // 8192*4096 = 33,554,432
  const long long totalGroups = n >> 2;             // groups of 4 elements
  const long long headGroups  = 4096 / 4;           // row 0 (H=4096) pass-through
  const long long ntiles      = totalGroups / GROUPS_PER_TILE;

  long long blocks64 = ntiles < 4096 ? (ntiles > 0 ? ntiles : 1) : 4096;
  sae_main<<<(int)blocks64, BLOCK_THREADS, 0, stream>>>(x, out, ntiles, headGroups);

  const long long tail = totalGroups - ntiles * GROUPS_PER_TILE;
  if (tail > 0) {
    const int tb = (int)((tail + BLOCK_THREADS - 1) / BLOCK_THREADS);
    sae_tail<<<tb, BLOCK_THREADS, 0, stream>>>(x, out, ntiles * GROUPS_PER_TILE,
                                               totalGroups, headGroups);
  }
}